// Experts_27745488732219
// MI455X (gfx1250) — compile-verified
//
#include <hip/hip_runtime.h>
#include <stdint.h>

// -----------------------------------------------------------------------------
// MoE grouped GEMM forward for MI455X (gfx1250, wave32, WMMA + async-to-LDS).
//   out[t, :] = X[t, :] @ W[e(t)] + bias[e(t)]
// T=8192, D=1024, O=4096, E=8.  All expert sizes are multiples of 128, so each
// 128-row output tile belongs to exactly one expert.
//
// Block tile 128x128.  K is processed in chunks of 16 that are streamed into
// double-buffered LDS with GLOBAL_LOAD_ASYNC_TO_LDS_B128 (ASYNCcnt pipeline),
// overlapping the next chunk's global traffic with the current chunk's WMMAs.
// 8 wave32 waves per block in a 2(M) x 4(N) arrangement; each wave computes a
// 64x32 region as 4x2 tiles of V_WMMA_F32_16X16X4_F32.
// -----------------------------------------------------------------------------

typedef __attribute__((ext_vector_type(2))) float v2f;
typedef __attribute__((ext_vector_type(8))) float v8f;

#define T_DIM 8192
#define D_DIM 1024
#define O_DIM 4096
#define E_NUM 8

#define BM 128
#define BN 128
#define KT 16
#define NKT (D_DIM / KT)  // 64 chunks
#define LDA 20    // A tile LDS dword stride: 20*i mod 64 distinct for i=0..15 (banks ≡0 mod 4)
#define LDB 136   // B tile LDS dword stride: +2 rows => +16 banks (disjoint half-wave windows)

// ---- CDNA5 async memory->LDS copy (tracked by ASYNCcnt) ---------------------
__device__ __forceinline__ void async_ld_b128(uint32_t lds_addr, const float* gaddr) {
    asm volatile("global_load_async_to_lds_b128 %0, %1, off"
                 :: "v"(lds_addr), "v"(gaddr) : "memory");
}
__device__ __forceinline__ void wait_async4() {
    asm volatile("s_wait_asynccnt 0x4" ::: "memory");
}
__device__ __forceinline__ void wait_async0() {
    asm volatile("s_wait_asynccnt 0x0" ::: "memory");
}

// Issue one K-chunk (A: BMxKT, B: KTxBN) into LDS; 4 async b128 per thread.
__device__ __forceinline__ void issue_tile(const float* __restrict__ Ag0,
                                           const float* __restrict__ Bg0,
                                           int kt, uint32_t a_lds, uint32_t b_lds,
                                           int tid) {
    const float* Ag = Ag0 + kt;                    // 128 rows x 16 cols (4 float4/row)
    #pragma unroll
    for (int i = 0; i < 2; ++i) {
        const int idx = tid + i * 256;             // 0..511
        const int r   = idx >> 2;
        const int c4  = (idx & 3) << 2;
        async_ld_b128(a_lds + (uint32_t)(r * LDA + c4) * 4u,
                      Ag + (size_t)r * D_DIM + c4);
    }
    const float* Bg = Bg0 + (size_t)kt * O_DIM;    // 16 rows x 128 cols (32 float4/row)
    #pragma unroll
    for (int i = 0; i < 2; ++i) {
        const int idx = tid + i * 256;             // 0..511
        const int r   = idx >> 5;
        const int c4  = (idx & 31) << 2;
        async_ld_b128(b_lds + (uint32_t)(r * LDB + c4) * 4u,
                      Bg + (size_t)r * O_DIM + c4);
    }
}

// Compute one K-chunk: 4 k-steps of K=4, 4x2 WMMA tiles per wave (32 WMMAs).
__device__ __forceinline__ void compute_chunk(const float* __restrict__ Ab,
                                              const float* __restrict__ Bb,
                                              int aoff, int boff, v8f acc[4][2]) {
    #pragma unroll
    for (int k0 = 0; k0 < KT; k0 += 4) {
        v2f a[4];
        v2f b[2];
        #pragma unroll
        for (int mi = 0; mi < 4; ++mi)
            a[mi] = *reinterpret_cast<const v2f*>(&Ab[aoff + mi * 16 * LDA + k0]);
        #pragma unroll
        for (int ni = 0; ni < 2; ++ni) {
            v2f t;
            t.x = Bb[boff + ni * 16 + k0 * LDB];
            t.y = Bb[boff + ni * 16 + (k0 + 1) * LDB];
            b[ni] = t;
        }
        #pragma unroll
        for (int mi = 0; mi < 4; ++mi)
            #pragma unroll
            for (int ni = 0; ni < 2; ++ni)
                acc[mi][ni] = __builtin_amdgcn_wmma_f32_16x16x4_f32(
                    /*neg_a=*/false, a[mi],
                    /*neg_b=*/false, b[ni],
                    /*c_mod=*/(short)0, acc[mi][ni],
                    /*reuse_a=*/false, /*reuse_b=*/false);
    }
}

__launch_bounds__(256, 2)
__global__ void moe_gemm_f32_wmma(const float* __restrict__ X,
                                  const float* __restrict__ W,
                                  const float* __restrict__ Bias,
                                  const int*   __restrict__ expert_size,
                                  float*       __restrict__ Out)
{
    __shared__ float Asm[2][BM * LDA];   // 2 x 10240 B
    __shared__ float Bsm[2][KT * LDB];   // 2 x  8704 B   (total 37888 B)

    const int tid     = threadIdx.x;
    const int lane    = tid & 31;
    const int wave    = tid >> 5;
    const int lane_lo = lane & 15;    // row/col within 16
    const int lane_hi = lane >> 4;    // K pair select (A/B), M+8 half (C/D)

    const int wave_m  = wave & 1;     // 0..1 -> 64 rows each
    const int wave_n  = wave >> 1;    // 0..3 -> 32 cols each

    const int tile_n  = blockIdx.x;   // O/128 = 32
    const int tile_m  = blockIdx.y;   // T/128 = 64
    const int row0    = tile_m * BM;
    const int col0    = tile_n * BN;

    // ---- which expert owns this row tile (sizes are multiples of 128) ----
    int e = 0;
    {
        int off = 0;
        #pragma unroll
        for (int i = 0; i < E_NUM; ++i) {
            int sz = expert_size[i];
            if (row0 >= off && row0 < off + sz) e = i;
            off += sz;
        }
    }

    const float* __restrict__ Ag0 = X + (size_t)row0 * D_DIM;
    const float* __restrict__ Bg0 = W + (size_t)e * (size_t)D_DIM * O_DIM + col0;

    // LDS byte addresses for the async DMA (flat addr[31:0] == LDS byte offset).
    const uint32_t a_lds0 = (uint32_t)(uintptr_t)&Asm[0][0];
    const uint32_t a_lds1 = (uint32_t)(uintptr_t)&Asm[1][0];
    const uint32_t b_lds0 = (uint32_t)(uintptr_t)&Bsm[0][0];
    const uint32_t b_lds1 = (uint32_t)(uintptr_t)&Bsm[1][0];

    // Per-wave fragment base offsets (dwords) inside the LDS tiles.
    const int aoff = (wave_m * 64 + lane_lo) * LDA + lane_hi * 2;
    const int boff = (lane_hi * 2) * LDB + wave_n * 32 + lane_lo;

    v8f acc[4][2];
    #pragma unroll
    for (int mi = 0; mi < 4; ++mi)
        #pragma unroll
        for (int ni = 0; ni < 2; ++ni)
            acc[mi][ni] = (v8f){0.f, 0.f, 0.f, 0.f, 0.f, 0.f, 0.f, 0.f};

    // ---- software pipeline: prefetch chunk c+1 while computing chunk c ----
    issue_tile(Ag0, Bg0, 0, a_lds0, b_lds0, tid);

    for (int c = 0; c < NKT; c += 2) {
        // phase 0: compute buffer 0, prefetch chunk c+1 into buffer 1
        issue_tile(Ag0, Bg0, (c + 1) * KT, a_lds1, b_lds1, tid);  // c+1 <= 63 always
        wait_async4();
        __syncthreads();
        compute_chunk(&Asm[0][0], &Bsm[0][0], aoff, boff, acc);
        __syncthreads();

        // phase 1: compute buffer 1, prefetch chunk c+2 into buffer 0
        if (c + 2 < NKT) {
            issue_tile(Ag0, Bg0, (c + 2) * KT, a_lds0, b_lds0, tid);
            wait_async4();
        } else {
            wait_async0();
        }
        __syncthreads();
        compute_chunk(&Asm[1][0], &Bsm[1][0], aoff, boff, acc);
        __syncthreads();
    }

    // ---- epilogue: add bias, store ----
    // C/D layout: VGPR r holds M = (lane_hi*8 + r), N = lane_lo within the tile.
    #pragma unroll
    for (int ni = 0; ni < 2; ++ni) {
        const int c  = col0 + wave_n * 32 + ni * 16 + lane_lo;
        const float bv = Bias[e * O_DIM + c];
        #pragma unroll
        for (int mi = 0; mi < 4; ++mi) {
            const int rbase = row0 + wave_m * 64 + mi * 16 + lane_hi * 8;
            #pragma unroll
            for (int r = 0; r < 8; ++r) {
                Out[(size_t)(rbase + r) * O_DIM + c] = acc[mi][ni][r] + bv;
            }
        }
    }
}

extern "C" void kernel_launch(void* const* d_in, const int* in_sizes, int n_in,
                              void* d_out, int out_size, void* d_ws, size_t ws_size,
                              hipStream_t stream)
{
    (void)in_sizes; (void)n_in; (void)out_size; (void)d_ws; (void)ws_size;

    const float* X    = (const float*)d_in[0];   // [T, D] f32
    const float* W    = (const float*)d_in[1];   // [E, D, O] f32
    const float* Bias = (const float*)d_in[2];   // [E, O] f32
    const int*   es   = (const int*)  d_in[3];   // [E] i32
    float*       out  = (float*)d_out;           // [T, O] f32

    dim3 grid(O_DIM / BN, T_DIM / BM);           // (32, 64)
    dim3 block(256);                             // 8 wave32 waves
    moe_gemm_f32_wmma<<<grid, block, 0, stream>>>(X, W, Bias, es, out);
}